// OneMatch_G_30932354466116
// MI455X (gfx1250) — compile-verified
//
#include <hip/hip_runtime.h>
#include <limits.h>

typedef __attribute__((ext_vector_type(2))) float v2f;
typedef __attribute__((ext_vector_type(8))) float v8f;

#define NB        4
#define L_DIM     4096
#define S_DIM     4096
#define C_DIM     256
#define HW        64
#define STRIDE_PX 8
// conf = (vi/sqrt(c)) . (ir/sqrt(c)) / TEMP == dot / (c * TEMP) = dot / 25.6
#define CONF_SCALE  (1.0f / (256.0f * 0.1f))
#define CONF_THRESH 0.2f

// Order-preserving float<->int bijection (signed-int order == float order).
__device__ __forceinline__ int f2oi(float f) {
    int i = __float_as_int(f);
    return i >= 0 ? i : (i ^ 0x7FFFFFFF);
}
__device__ __forceinline__ float oi2f(int i) {
    return __int_as_float(i >= 0 ? i : (i ^ 0x7FFFFFFF));
}

// 64-bit key: high = unsigned-ordered float encoding, low = ~s.
// atomicMax(u64) => max value, ties broken toward the SMALLEST index s
// (reproduces torch argmax-of-bool "first True" semantics).
__device__ __forceinline__ unsigned long long packKey(float v, int s) {
    unsigned int uo = (unsigned int)f2oi(v) ^ 0x80000000u;
    return ((unsigned long long)uo << 32) | (unsigned int)(~s);
}
__device__ __forceinline__ float keyVal(unsigned long long k) {
    return oi2f((int)((unsigned int)(k >> 32) ^ 0x80000000u));
}
__device__ __forceinline__ int keyIdx(unsigned long long k) {
    return (int)(~(unsigned int)k);
}

__global__ void init_ws_kernel(unsigned long long* __restrict__ rowBest,
                               int* __restrict__ colMax) {
    int i = blockIdx.x * blockDim.x + threadIdx.x;
    if (i < NB * L_DIM) {
        rowBest[i] = 0ull;            // below any real key
        colMax[i]  = (int)0x80000000; // -inf in ordered-int space
    }
}

// Single fused GEMM pass: conf tiles stay in accumulator VGPRs; commit
// per-row (value, argmin-index) keys and per-column max values.
__global__ __launch_bounds__(128) void conf_gemm_kernel(
    const float* __restrict__ vi, const float* __restrict__ ir,
    unsigned long long* __restrict__ rowBest, int* __restrict__ colMax) {
    const int n    = blockIdx.z;
    const int lane = threadIdx.x & 31;
    const int wave = threadIdx.x >> 5;
    const int m16  = lane & 15;   // row-in-tile for A frag, col-in-tile for B frag / C
    const int kh   = lane >> 4;   // which K-half of the x4 step this lane holds

    const int l0 = blockIdx.y * 128 + (wave >> 1) * 64;
    const int s0 = blockIdx.x * 128 + (wave & 1) * 64;

    const float* __restrict__ vin = vi + (size_t)n * L_DIM * C_DIM;
    const float* __restrict__ irn = ir + (size_t)n * S_DIM * C_DIM;

    // A frag (16x4 f32): lane = m + 16*kh holds A[m][k+2kh], A[m][k+2kh+1]
    // B frag (4x16 f32): lane = n + 16*kh holds B[k+2kh][n], B[k+2kh+1][n]
    //   with B[k'][n] = ir[s0+n][k'] -> also a contiguous 2-float row load.
    const float* aRow[4];
    const float* bRow[4];
#pragma unroll
    for (int t = 0; t < 4; ++t) {
        aRow[t] = vin + (size_t)(l0 + 16 * t + m16) * C_DIM + 2 * kh;
        bRow[t] = irn + (size_t)(s0 + 16 * t + m16) * C_DIM + 2 * kh;
    }

    v8f acc[4][4];
    const v8f zero = {0.f, 0.f, 0.f, 0.f, 0.f, 0.f, 0.f, 0.f};
#pragma unroll
    for (int i = 0; i < 4; ++i)
#pragma unroll
        for (int j = 0; j < 4; ++j) acc[i][j] = zero;

    for (int k = 0; k < C_DIM; k += 4) {
        v2f a[4], b[4];
#pragma unroll
        for (int t = 0; t < 4; ++t) {
            a[t] = *(const v2f*)(aRow[t] + k);
            b[t] = *(const v2f*)(bRow[t] + k);
        }
#pragma unroll
        for (int i = 0; i < 4; ++i)
#pragma unroll
            for (int j = 0; j < 4; ++j)
                acc[i][j] = __builtin_amdgcn_wmma_f32_16x16x4_f32(
                    false, a[i], false, b[j], (short)0, acc[i][j], false, false);
    }

#pragma unroll
    for (int i = 0; i < 4; ++i)
#pragma unroll
        for (int j = 0; j < 4; ++j)
#pragma unroll
            for (int v = 0; v < 8; ++v) acc[i][j][v] *= CONF_SCALE;

    // C layout: VGPR v, lanes 0-15  -> (M = v,     N = lane)
    //                    lanes 16-31 -> (M = v + 8, N = lane - 16)

    // ---- per-row best (value, min index) over this wave's 64 columns ----
#pragma unroll
    for (int i = 0; i < 4; ++i) {
#pragma unroll
        for (int v = 0; v < 8; ++v) {
            unsigned long long key = packKey(acc[i][0][v], s0 + m16);
#pragma unroll
            for (int j = 1; j < 4; ++j) {
                unsigned long long kj = packKey(acc[i][j][v], s0 + 16 * j + m16);
                key = key > kj ? key : kj;
            }
            // butterfly over the 16-lane group (covers all 64 columns)
#pragma unroll
            for (int d = 1; d <= 8; d <<= 1) {
                unsigned long long o = __shfl_xor(key, d, 32);
                key = key > o ? key : o;
            }
            if (m16 == v)
                atomicMax(&rowBest[n * L_DIM + l0 + 16 * i + v + 8 * kh], key);
        }
    }

    // ---- per-column max value over this wave's 64 rows ----
#pragma unroll
    for (int j = 0; j < 4; ++j) {
        v8f cm = acc[0][j];
#pragma unroll
        for (int i = 1; i < 4; ++i)
#pragma unroll
            for (int v = 0; v < 8; ++v) cm[v] = fmaxf(cm[v], acc[i][j][v]);
        float c = cm[0];
#pragma unroll
        for (int v = 1; v < 8; ++v) c = fmaxf(c, cm[v]);
        c = fmaxf(c, __shfl_xor(c, 16, 32));  // lanes L and L^16 hold same column
        if (lane < 16)
            atomicMax(&colMax[n * S_DIM + s0 + 16 * j + m16], f2oi(c));
    }
}

// Mutual-NN check + output formatting: O(NB*L) threads.
__global__ void finalize_kernel(const unsigned long long* __restrict__ rowBest,
                                const int* __restrict__ colMax,
                                float* __restrict__ out) {
    int i = blockIdx.x * blockDim.x + threadIdx.x;
    if (i >= NB * L_DIM) return;
    int n = i / L_DIM;
    int l = i % L_DIM;

    unsigned long long key = rowBest[i];
    float vr   = keyVal(key);
    int   sstr = keyIdx(key) & (S_DIM - 1);

    float cm = oi2f(colMax[n * S_DIM + sstr]);
    bool valid = (vr > CONF_THRESH) && (vr == cm);
    int j = valid ? sstr : 0;

    float* mk0 = out;                  // [NB, L, 2]
    float* mk1 = out + NB * L_DIM * 2; // [NB, L, 2]
    float* val = out + NB * L_DIM * 4; // [NB, L]

    mk0[2 * i + 0] = (float)((l % HW) * STRIDE_PX);
    mk0[2 * i + 1] = (float)((l / HW) * STRIDE_PX);
    mk1[2 * i + 0] = (float)((j % HW) * STRIDE_PX);
    mk1[2 * i + 1] = (float)((j / HW) * STRIDE_PX);
    val[i] = valid ? 1.0f : 0.0f;
}

extern "C" void kernel_launch(void* const* d_in, const int* in_sizes, int n_in,
                              void* d_out, int out_size, void* d_ws, size_t ws_size,
                              hipStream_t stream) {
    const float* vi = (const float*)d_in[0];
    const float* ir = (const float*)d_in[1];
    float* out = (float*)d_out;

    unsigned long long* rowBest = (unsigned long long*)d_ws;   // NB*L_DIM u64
    int* colMax = (int*)(rowBest + NB * L_DIM);                // NB*S_DIM int

    dim3 gGemm(S_DIM / 128, L_DIM / 128, NB);
    dim3 bGemm(128);

    init_ws_kernel<<<(NB * L_DIM + 255) / 256, 256, 0, stream>>>(rowBest, colMax);
    conf_gemm_kernel<<<gGemm, bGemm, 0, stream>>>(vi, ir, rowBest, colMax);
    finalize_kernel<<<(NB * L_DIM + 255) / 256, 256, 0, stream>>>(rowBest, colMax, out);
}